// TM_HGNN_69861938036847
// MI455X (gfx1250) — compile-verified
//
#include <hip/hip_runtime.h>

#define META 4
#define EMB 768
#define PROJ 256
#define HID 256
#define K1 (META + PROJ)   // 260
#define G_GRAPHS 64
#define BN_EPS 1e-5f

typedef __attribute__((ext_vector_type(2))) float v2f;
typedef __attribute__((ext_vector_type(8))) float v8f;

// ---------------------------------------------------------------------------
// C[M x 256] = A[M x K] @ B[K x 256] (+bias), all row-major fp32.
// A addressed as A[row*lda + aoff + k]; C as C[row*ldc + coff + col].
// Block = 256 threads (8 waves), tile = 128 rows x 64 cols, K chunked by 64
// through LDS. Each wave owns 16 rows x 64 cols = 4 x v_wmma_f32_16x16x4_f32
// accumulators (exact fp32 — matches the fp32 reference bit-for-bit in math).
// ---------------------------------------------------------------------------
__global__ __launch_bounds__(256)
void wmma_gemm_f32(const float* __restrict__ A, int lda, int aoff,
                   const float* __restrict__ B,
                   const float* __restrict__ bias,
                   float* __restrict__ C, int ldc, int coff,
                   int M, int K)
{
    __shared__ float As[128][68];   // pad 68: conflict-free for both lane halves
    __shared__ float Bs[64][68];

    const int tid  = threadIdx.x;
    const int wave = tid >> 5;          // 0..7
    const int lane = tid & 31;
    const int m16  = lane & 15;
    const int kh   = lane >> 4;         // 0/1: which K/row half this lane holds
    const int rowBase = blockIdx.x * 128;
    const int colBase = blockIdx.y * 64;

    v8f acc[4] = {};                    // 4 n-tiles of 16 cols each

    for (int k0 = 0; k0 < K; k0 += 64) {
        const int kc = (K - k0) < 64 ? (K - k0) : 64;   // always multiple of 4 here

        // Stage A chunk (coalesced: consecutive tid -> consecutive k)
        for (int idx = tid; idx < 128 * 64; idx += 256) {
            int r = idx >> 6, kk = idx & 63;
            int row = rowBase + r;
            As[r][kk] = (row < M && kk < kc)
                      ? A[(size_t)row * lda + aoff + k0 + kk] : 0.0f;
        }
        // Stage B chunk (coalesced: consecutive tid -> consecutive col)
        for (int idx = tid; idx < 64 * 64; idx += 256) {
            int kk = idx >> 6, c = idx & 63;
            Bs[kk][c] = (kk < kc)
                      ? B[(size_t)(k0 + kk) * 256 + colBase + c] : 0.0f;
        }
        __syncthreads();

        for (int kk = 0; kk < kc; kk += 4) {
            // A 16x4 operand: lane(m,kh) holds A[m][2*kh+j]
            v2f a;
            a.x = As[wave * 16 + m16][kk + 2 * kh + 0];
            a.y = As[wave * 16 + m16][kk + 2 * kh + 1];
#pragma unroll
            for (int t = 0; t < 4; ++t) {
                // B 4x16 operand: lane(n,kh) holds B[2*kh+j][n]
                v2f b;
                b.x = Bs[kk + 2 * kh + 0][t * 16 + m16];
                b.y = Bs[kk + 2 * kh + 1][t * 16 + m16];
                acc[t] = __builtin_amdgcn_wmma_f32_16x16x4_f32(
                    false, a, false, b, (short)0, acc[t], false, false);
            }
        }
        __syncthreads();
    }

    // C/D layout: VGPR r holds row r (lanes 0-15) and row r+8 (lanes 16-31)
#pragma unroll
    for (int t = 0; t < 4; ++t) {
#pragma unroll
        for (int r = 0; r < 8; ++r) {
            int row = rowBase + wave * 16 + r + 8 * kh;
            int col = colBase + t * 16 + m16;
            if (row < M) {
                float v = acc[t][r];
                if (bias) v += bias[col];
                C[(size_t)row * ldc + coff + col] = v;
            }
        }
    }
}

// --- degree accumulation (self-loop +1 folded into inv kernel) --------------
__global__ void degrees_kernel(const int* __restrict__ dst, const int* __restrict__ mask,
                               float* d1, float* d2, float* d3, int E)
{
    int e = blockIdx.x * 256 + threadIdx.x;
    if (e >= E) return;
    int d = dst[e];
    atomicAdd(&d1[d], 1.0f);
    int m = mask[e];
    if (m == 1)      atomicAdd(&d2[d], 1.0f);
    else if (m == 2) atomicAdd(&d3[d], 1.0f);
}

__global__ void invdeg_kernel(float* d1, float* d2, float* d3, int N)
{
    int i = blockIdx.x * 256 + threadIdx.x;
    if (i >= N) return;
    d1[i] = rsqrtf(d1[i] + 1.0f);   // deg always > 0 after self-loop
    d2[i] = rsqrtf(d2[i] + 1.0f);
    d3[i] = rsqrtf(d3[i] + 1.0f);
}

__global__ void copy_meta_kernel(const float* __restrict__ x, float* __restrict__ h0, int N)
{
    int idx = blockIdx.x * 256 + threadIdx.x;
    if (idx >= N * META) return;
    int n = idx >> 2, j = idx & 3;
    h0[(size_t)n * K1 + j] = x[(size_t)n * (META + EMB) + j];
}

// --- edge scatter: agg[dst] += hw[src] * inv[src]*inv[dst], mask-filtered ---
__global__ __launch_bounds__(256)
void edge_scatter_kernel(const float* __restrict__ hw, float* __restrict__ agg,
                         const int* __restrict__ src, const int* __restrict__ dst,
                         const int* __restrict__ mask, int maskval,
                         const float* __restrict__ inv, int E)
{
    int gid = blockIdx.x * 256 + threadIdx.x;
    int e = gid >> 6;        // 64 threads per edge, 4 floats each
    int t = gid & 63;
    if (e >= E) return;
    if (maskval >= 0 && mask[e] != maskval) return;
    int s = src[e], d = dst[e];
    float coef = inv[s] * inv[d];
    const float4 v = *(const float4*)(hw + (size_t)s * HID + t * 4);
    float* a = agg + (size_t)d * HID + t * 4;
    atomicAdd(a + 0, v.x * coef);
    atomicAdd(a + 1, v.y * coef);
    atomicAdd(a + 2, v.z * coef);
    atomicAdd(a + 3, v.w * coef);
}

// --- pre = agg + hw*inv^2 + b (in place in agg) + per-feature BN stats ------
__global__ __launch_bounds__(256)
void combine_stats_kernel(const float* __restrict__ hw, float* __restrict__ agg,
                          const float* __restrict__ inv, const float* __restrict__ bias,
                          float* __restrict__ stat, int N)
{
    int f = threadIdx.x;            // feature 0..255
    int base = blockIdx.x * 64;
    float s = 0.0f, sq = 0.0f;
    for (int r = 0; r < 64; ++r) {
        int i = base + r;
        if (i >= N) break;
        float iv = inv[i];
        size_t o = (size_t)i * HID + f;
        float pre = agg[o] + hw[o] * iv * iv + bias[f];
        agg[o] = pre;
        s += pre; sq += pre * pre;
    }
    atomicAdd(&stat[f], s);
    atomicAdd(&stat[256 + f], sq);
}

__global__ void bn_finalize_kernel(const float* __restrict__ stat,
                                   float* __restrict__ mu, float* __restrict__ rsig, float n)
{
    int f = threadIdx.x;
    float m = stat[f] / n;
    float v = stat[256 + f] / n - m * m;
    mu[f] = m;
    rsig[f] = rsqrtf(v + BN_EPS);
}

__global__ void bn_apply_relu_kernel(const float* __restrict__ pre, float* __restrict__ out,
                                     const float* __restrict__ mu, const float* __restrict__ rsig,
                                     const float* __restrict__ g, const float* __restrict__ be, int N)
{
    int i = blockIdx.x, f = threadIdx.x;
    size_t o = (size_t)i * HID + f;
    float v = (pre[o] - mu[f]) * rsig[f] * g[f] + be[f];
    out[o] = fmaxf(v, 0.0f);
}

// --- mean pool: batch is sorted, so flush one atomic per run per block ------
__global__ __launch_bounds__(256)
void pool_sum_kernel(const float* __restrict__ h, const int* __restrict__ batch,
                     float* __restrict__ pool, int N)
{
    int f = threadIdx.x;
    int base = blockIdx.x * 64;
    int gcur = -1; float acc = 0.0f;
    for (int r = 0; r < 64; ++r) {
        int i = base + r;
        if (i >= N) break;
        int g = batch[i];
        if (g != gcur) {
            if (gcur >= 0) atomicAdd(&pool[(size_t)gcur * HID + f], acc);
            acc = 0.0f; gcur = g;
        }
        acc += h[(size_t)i * HID + f];
    }
    if (gcur >= 0) atomicAdd(&pool[(size_t)gcur * HID + f], acc);
}

__global__ void pool_cnt_kernel(const int* __restrict__ batch, float* __restrict__ cnt, int N)
{
    int i = blockIdx.x * 256 + threadIdx.x;
    if (i >= N) return;
    atomicAdd(&cnt[batch[i]], 1.0f);
}

__global__ void final_out_kernel(const float* __restrict__ pool, const float* __restrict__ cnt,
                                 const float* __restrict__ Wl, const float* __restrict__ bl,
                                 float* __restrict__ out)
{
    __shared__ float red[256];
    int g = blockIdx.x, f = threadIdx.x;
    float c = fmaxf(cnt[g], 1.0f);
    red[f] = (pool[(size_t)g * HID + f] / c) * Wl[f];
    __syncthreads();
    for (int s = 128; s > 0; s >>= 1) {
        if (f < s) red[f] += red[f + s];
        __syncthreads();
    }
    if (f == 0) out[g] = red[0] + bl[0];
}

// ---------------------------------------------------------------------------
extern "C" void kernel_launch(void* const* d_in, const int* in_sizes, int n_in,
                              void* d_out, int out_size, void* d_ws, size_t ws_size,
                              hipStream_t stream)
{
    const float* x   = (const float*)d_in[0];
    const int*   ei  = (const int*)d_in[1];
    const int*   em  = (const int*)d_in[2];
    const int*   bat = (const int*)d_in[3];
    const float* Wp  = (const float*)d_in[4];
    const float* bp  = (const float*)d_in[5];
    const float* W1  = (const float*)d_in[6];
    const float* b1  = (const float*)d_in[7];
    const float* W2  = (const float*)d_in[8];
    const float* b2  = (const float*)d_in[9];
    const float* W3  = (const float*)d_in[10];
    const float* b3  = (const float*)d_in[11];
    const float* g1  = (const float*)d_in[12]; const float* be1 = (const float*)d_in[13];
    const float* g2  = (const float*)d_in[14]; const float* be2 = (const float*)d_in[15];
    const float* g3  = (const float*)d_in[16]; const float* be3 = (const float*)d_in[17];
    const float* Wl  = (const float*)d_in[18]; const float* bl  = (const float*)d_in[19];
    float* out = (float*)d_out;

    const int N = in_sizes[0] / (META + EMB);
    const int E = in_sizes[1] / 2;
    const int* src = ei;
    const int* dst = ei + E;

    float* ws   = (float*)d_ws;
    float* h0   = ws;                          // N x 260 (features; stride 260 then 256)
    float* hw   = h0  + (size_t)N * K1;        // N x 256 (h @ W)
    float* agg  = hw  + (size_t)N * HID;       // N x 256 (agg, then pre in place)
    float* inv1 = agg + (size_t)N * HID;       // N (deg -> inv)
    float* inv2 = inv1 + N;
    float* inv3 = inv2 + N;
    float* stat = inv3 + N;                    // 512 (sum, sumsq)
    float* mu   = stat + 512;                  // 256
    float* rsig = mu   + 256;                  // 256
    float* pool = rsig + 256;                  // 64 x 256
    float* cnt  = pool + (size_t)G_GRAPHS * HID; // 64

    // degrees -> inv (self-loop folded in)
    hipMemsetAsync(inv1, 0, sizeof(float) * 3 * (size_t)N, stream);
    degrees_kernel<<<(E + 255) / 256, 256, 0, stream>>>(dst, em, inv1, inv2, inv3, E);
    invdeg_kernel<<<(N + 255) / 256, 256, 0, stream>>>(inv1, inv2, inv3, N);

    // h0 = [x_meta | x_emb @ Wp + bp]
    copy_meta_kernel<<<(N * META + 255) / 256, 256, 0, stream>>>(x, h0, N);
    dim3 ggrid((N + 127) / 128, 4);
    wmma_gemm_f32<<<ggrid, 256, 0, stream>>>(x, META + EMB, META, Wp, bp,
                                             h0, K1, META, N, EMB);

    struct Layer { const float *W, *b, *g, *be, *inv; int mv, K; };
    Layer L[3] = {
        { W1, b1, g1, be1, inv1, -1, K1  },
        { W2, b2, g2, be2, inv2,  1, HID },
        { W3, b3, g3, be3, inv3,  2, HID },
    };

    for (int c = 0; c < 3; ++c) {
        // hw = h_in @ Wc     (h_in lives in h0: stride 260 for c==0, else 256)
        wmma_gemm_f32<<<ggrid, 256, 0, stream>>>(h0, L[c].K, 0, L[c].W, nullptr,
                                                 hw, HID, 0, N, L[c].K);
        hipMemsetAsync(agg, 0, sizeof(float) * (size_t)N * HID, stream);
        hipMemsetAsync(stat, 0, sizeof(float) * 512, stream);
        edge_scatter_kernel<<<(E * 64 + 255) / 256, 256, 0, stream>>>(
            hw, agg, src, dst, em, L[c].mv, L[c].inv, E);
        combine_stats_kernel<<<(N + 63) / 64, 256, 0, stream>>>(
            hw, agg, L[c].inv, L[c].b, stat, N);
        bn_finalize_kernel<<<1, 256, 0, stream>>>(stat, mu, rsig, (float)N);
        bn_apply_relu_kernel<<<N, 256, 0, stream>>>(agg, h0, mu, rsig, L[c].g, L[c].be, N);
    }

    // mean pool + final linear
    hipMemsetAsync(pool, 0, sizeof(float) * ((size_t)G_GRAPHS * HID + G_GRAPHS), stream);
    pool_sum_kernel<<<(N + 63) / 64, 256, 0, stream>>>(h0, bat, pool, N);
    pool_cnt_kernel<<<(N + 255) / 256, 256, 0, stream>>>(bat, cnt, N);
    final_out_kernel<<<G_GRAPHS, 256, 0, stream>>>(pool, cnt, Wl, bl, out);
}